// RealQuantizerBase_46815143526604
// MI455X (gfx1250) — compile-verified
//
#include <hip/hip_runtime.h>

// uint4 group-dequant:  out[n,k] = (q - zp[n,g]) * scale_t[g,n]
// N = K = 8192, G = 128, NG = 64, 8 nibbles per packed int32 (low nibble first).
//
// Store-bound streaming kernel (~290 MB moved, ~12.5 us at 23.3 TB/s).
// gfx1250 path: async global->LDS staging (ASYNCcnt) + NT b128 stores.

typedef float v4f __attribute__((ext_vector_type(4)));
typedef int   v4i __attribute__((ext_vector_type(4)));

#define N_ROWS 8192
#define K_COLS 8192
#define PACKED_PER_ROW (K_COLS / 8) /* 1024 int32 per row */
#define ZPW_PER_ROW 8               /* 64 groups / 8 nibbles */

__global__ __launch_bounds__(256) void dequant_u4_rows(
    const int* __restrict__ x_packed,   // [N, 1024] int32
    const float* __restrict__ scale_t,  // [64, N]   f32 (transposed)
    const int* __restrict__ zp_packed,  // [N, 8]    int32
    float* __restrict__ out)            // [N, 8192] f32
{
  __shared__ int lds_x[PACKED_PER_ROW]; // 4 KB: one row of packed weights

  const int n = blockIdx.x;
  const int t = threadIdx.x;

  // ---- async stage: 256 threads x b128 = 4 KB row into LDS ----------------
  const int* gsrc = x_packed + (size_t)n * PACKED_PER_ROW + t * 4;
#if __has_builtin(__builtin_amdgcn_global_load_async_to_lds_b128)
  __builtin_amdgcn_global_load_async_to_lds_b128(
      (v4i*)gsrc, (v4i*)&lds_x[t * 4], 0, 0);
#if __has_builtin(__builtin_amdgcn_s_wait_asynccnt)
  __builtin_amdgcn_s_wait_asynccnt(0);
#else
  asm volatile("s_wait_asynccnt 0" ::: "memory");
#endif
#else
  // Fallback: synchronous b128 stage through VGPRs.
  *(v4i*)&lds_x[t * 4] = *(const v4i*)gsrc;
#endif
  __syncthreads();

  // Wave-uniform group id component: g = 8*k + wid  (wid = wave id in block).
  const int wid = __builtin_amdgcn_readfirstlane(t >> 5); // 0..7, SGPR
  const int sh0 = (t & 1) << 4; // which half-dword of the packed word

  const float* srow = scale_t + n;                    // + g*N per group
  const int* zrow = zp_packed + (size_t)n * ZPW_PER_ROW;
  float* orow = out + (size_t)n * K_COLS;

#pragma unroll
  for (int k = 0; k < 8; ++k) {
    const int g = k * 8 + wid;                        // scalar
    const float s = srow[(size_t)g * N_ROWS];         // s_load (uniform)
    const int zw = zrow[k];                           // s_load (uniform)
    const float c = -(float)((zw >> (wid * 4)) & 0xF) * s; // c = -zp*s

    // element base e0 = k*1024 + 4t; packed word = e0/8 = k*128 + t/2
    const int w = lds_x[k * 128 + (t >> 1)];          // conflict-free ds read

    v4f o;
    o.x = __builtin_fmaf((float)((w >> (sh0 + 0)) & 0xF), s, c);
    o.y = __builtin_fmaf((float)((w >> (sh0 + 4)) & 0xF), s, c);
    o.z = __builtin_fmaf((float)((w >> (sh0 + 8)) & 0xF), s, c);
    o.w = __builtin_fmaf((float)((w >> (sh0 + 12)) & 0xF), s, c);

    // lane-contiguous b128 NT store: 512 B per wave per instruction
    __builtin_nontemporal_store(o, (v4f*)(orow + k * 1024 + t * 4));
  }
}

extern "C" void kernel_launch(void* const* d_in, const int* in_sizes, int n_in,
                              void* d_out, int out_size, void* d_ws, size_t ws_size,
                              hipStream_t stream) {
  (void)in_sizes; (void)n_in; (void)out_size; (void)d_ws; (void)ws_size;
  const int* x_packed = (const int*)d_in[0];
  const float* scale_t = (const float*)d_in[1];
  const int* zp_packed = (const int*)d_in[2];
  float* out = (float*)d_out;

  dequant_u4_rows<<<N_ROWS, 256, 0, stream>>>(x_packed, scale_t, zp_packed, out);
}